// HSICLoss_19516331393396
// MI455X (gfx1250) — compile-verified
//
#include <hip/hip_runtime.h>

typedef __attribute__((ext_vector_type(2))) float v2f;
typedef __attribute__((ext_vector_type(8))) float v8f;

#define HN 4096
#define HD 64
#define HL 16

// ---------------------------------------------------------------------------
// Prep: transpose (N, D, L) -> (L, N, D) with a WMMA-fragment-major K
// permutation inside each row, SCALED by sqrt(2) so the Gram WMMA directly
// produces 2*xi.xj. Also computes unscaled squared row norms.
// Permutation: new[g*32 + c*2 + pos] = old[4c + 2g + pos], g in {0,1}.
// ---------------------------------------------------------------------------
__global__ void __launch_bounds__(256) hsic_prep(const float* __restrict__ src,
                                                 float* __restrict__ dst,
                                                 float* __restrict__ sq) {
  int tid = blockIdx.x * blockDim.x + threadIdx.x;   // 0 .. N*L-1
  int t = tid & (HL - 1);
  int i = tid >> 4;
  const float* s = src + (size_t)i * HD * HL + t;
  float* d = dst + (size_t)t * HN * HD + (size_t)i * HD;
  const float r2 = 1.41421356237309515f;  // sqrt(2)
  float acc = 0.f;
#pragma unroll
  for (int j = 0; j < 16; ++j) {
    const int g = j >> 3;
    const int m = j & 7;
    const int base = 8 * m + 2 * g;
    float4 v;
    v.x = s[(base + 0) * HL];
    v.y = s[(base + 1) * HL];
    v.z = s[(base + 4) * HL];
    v.w = s[(base + 5) * HL];
    acc += v.x * v.x + v.y * v.y + v.z * v.z + v.w * v.w;
    float4 o;
    o.x = v.x * r2; o.y = v.y * r2; o.z = v.z * r2; o.w = v.w * r2;
    *(float4*)(d + 4 * j) = o;
  }
  sq[t * HN + i] = acc;
}

// ---------------------------------------------------------------------------
// Main: fully-fused  -d2 = 2*xi.xj - sqi - sqj  via 17 K-chunks of
// WMMA f32 16x16x4 (chunk 16 carries the norm terms), then RBF + HSIC
// partials. Block = (16-row tile, timestep); 8 waves stride column tiles.
// Per block writes partials: [S, ax.ay, sum(ax), sum(ay)].
// ---------------------------------------------------------------------------
__global__ void __launch_bounds__(256) hsic_main(const float* __restrict__ xt,
                                                 const float* __restrict__ yt,
                                                 const float* __restrict__ sqx,
                                                 const float* __restrict__ sqy,
                                                 float* __restrict__ part) {
  const int t = blockIdx.y;
  const int bi = blockIdx.x << 4;
  const int w = threadIdx.x >> 5;       // wave 0..7
  const int lane = threadIdx.x & 31;
  const int lmod = lane & 15;           // N / M-within-half index
  const int lhi = lane >> 4;            // half-wave select
  const float hmask = (lhi == 0) ? 1.f : 0.f;

  const float* xs = xt + (size_t)t * HN * HD;
  const float* ys = yt + (size_t)t * HN * HD;
  const float* sx = sqx + t * HN;
  const float* sy = sqy + t * HN;

  // Resident A fragments for rows bi..bi+15 of x and y (permuted layout:
  // chunk c at row*64 + lhi*32 + 2c), plus the norm-carrying chunk 16:
  // A_extra row m = [-sq_m, 1, 0, 0]  (half 0 holds K0,K1; half 1 zeros).
  v2f axf[17], ayf[17];
  {
    const float* xr = xs + (size_t)(bi + lmod) * HD + lhi * 32;
    const float* yr = ys + (size_t)(bi + lmod) * HD + lhi * 32;
#pragma unroll
    for (int c = 0; c < 16; ++c) {
      axf[c] = *(const v2f*)(xr + 2 * c);
      ayf[c] = *(const v2f*)(yr + 2 * c);
    }
    axf[16] = (v2f){-sx[bi + lmod] * hmask, hmask};
    ayf[16] = (v2f){-sy[bi + lmod] * hmask, hmask};
  }

  float rsx[8], rsy[8];
#pragma unroll
  for (int v = 0; v < 8; ++v) { rsx[v] = 0.f; rsy[v] = 0.f; }
  float sacc = 0.f;

  for (int bjB = w; bjB < 256; bjB += 8) {   // uniform trip count: EXEC full
    const int bj = bjB << 4;
    // B_extra col n = [1, -sq_n, 0, 0]^T (half 0 holds K0,K1; half 1 zeros).
    const v2f bxe = {hmask, -sx[bj + lmod] * hmask};
    const v2f bye = {hmask, -sy[bj + lmod] * hmask};
    v8f accx = __builtin_amdgcn_wmma_f32_16x16x4_f32(false, axf[16], false, bxe,
                                                     (short)0, (v8f){}, false,
                                                     false);
    v8f accy = __builtin_amdgcn_wmma_f32_16x16x4_f32(false, ayf[16], false, bye,
                                                     (short)0, (v8f){}, false,
                                                     false);
    const float* bxp = xs + (size_t)(bj + lmod) * HD + lhi * 32;
    const float* byp = ys + (size_t)(bj + lmod) * HD + lhi * 32;
    // Two halves of 8 K-chunks: contiguous loads (merge to b128), then
    // interleaved x/y WMMA chains (independent accumulators).
#pragma unroll
    for (int h = 0; h < 2; ++h) {
      v2f bx[8], by[8];
#pragma unroll
      for (int q = 0; q < 8; ++q) {
        bx[q] = *(const v2f*)(bxp + h * 16 + 2 * q);
        by[q] = *(const v2f*)(byp + h * 16 + 2 * q);
      }
#pragma unroll
      for (int q = 0; q < 8; ++q) {
        accx = __builtin_amdgcn_wmma_f32_16x16x4_f32(false, axf[h * 8 + q], false,
                                                     bx[q], (short)0, accx, false,
                                                     false);
        accy = __builtin_amdgcn_wmma_f32_16x16x4_f32(false, ayf[h * 8 + q], false,
                                                     by[q], (short)0, accy, false,
                                                     false);
      }
    }
    // acc = -d2 ; clamp d2>=0  <=>  min(acc, 0) ; K = exp(0.5 * acc).
#pragma unroll
    for (int v = 0; v < 8; ++v) {
      float kx = __expf(0.5f * fminf(accx[v], 0.f));
      float ky = __expf(0.5f * fminf(accy[v], 0.f));
      sacc = fmaf(kx, ky, sacc);
      rsx[v] += kx;
      rsy[v] += ky;
    }
  }

  // Reduce row sums across the 16-lane half-group (xor<16 keeps half fixed).
#pragma unroll
  for (int v = 0; v < 8; ++v) {
#pragma unroll
    for (int off = 8; off >= 1; off >>= 1) {
      rsx[v] += __shfl_xor(rsx[v], off, 32);
      rsy[v] += __shfl_xor(rsy[v], off, 32);
    }
  }
#pragma unroll
  for (int off = 16; off >= 1; off >>= 1) sacc += __shfl_xor(sacc, off, 32);

  __shared__ float lds_ax[8][16];
  __shared__ float lds_ay[8][16];
  __shared__ float lds_s[8];
  if (lmod == 0) {
#pragma unroll
    for (int v = 0; v < 8; ++v) {
      lds_ax[w][v + 8 * lhi] = rsx[v];
      lds_ay[w][v + 8 * lhi] = rsy[v];
    }
    if (lane == 0) lds_s[w] = sacc;
  }
  __syncthreads();

  if (threadIdx.x < 16) {
    int r = threadIdx.x;
    float axr = 0.f, ayr = 0.f;
#pragma unroll
    for (int ww = 0; ww < 8; ++ww) {
      axr += lds_ax[ww][r];
      ayr += lds_ay[ww][r];
    }
    float dt = axr * ayr;
#pragma unroll
    for (int off = 8; off >= 1; off >>= 1) {
      axr += __shfl_xor(axr, off, 32);
      ayr += __shfl_xor(ayr, off, 32);
      dt += __shfl_xor(dt, off, 32);
    }
    if (r == 0) {
      float S = 0.f;
#pragma unroll
      for (int ww = 0; ww < 8; ++ww) S += lds_s[ww];
      float* p = part + ((size_t)t * 256 + blockIdx.x) * 4;
      p[0] = S;
      p[1] = dt;
      p[2] = axr;
      p[3] = ayr;
    }
  }
}

// ---------------------------------------------------------------------------
// Finalize: reduce 256 block partials per timestep, apply HSIC combination,
// sum over timesteps. Deterministic (fixed reduction order, no atomics).
// ---------------------------------------------------------------------------
__global__ void __launch_bounds__(256) hsic_final(const float* __restrict__ part,
                                                  float* __restrict__ out) {
  __shared__ float red[4][8];
  const int lane = threadIdx.x & 31;
  const int w = threadIdx.x >> 5;
  const float n = (float)HN;
  float total = 0.f;
  for (int t = 0; t < HL; ++t) {
    const float* p = part + ((size_t)t * 256 + threadIdx.x) * 4;
    float S = p[0], dt = p[1], tx = p[2], ty = p[3];
#pragma unroll
    for (int off = 16; off >= 1; off >>= 1) {
      S += __shfl_xor(S, off, 32);
      dt += __shfl_xor(dt, off, 32);
      tx += __shfl_xor(tx, off, 32);
      ty += __shfl_xor(ty, off, 32);
    }
    if (lane == 0) { red[0][w] = S; red[1][w] = dt; red[2][w] = tx; red[3][w] = ty; }
    __syncthreads();
    if (threadIdx.x == 0) {
      float Ss = 0.f, ds = 0.f, txs = 0.f, tys = 0.f;
#pragma unroll
      for (int ww = 0; ww < 8; ++ww) {
        Ss += red[0][ww]; ds += red[1][ww]; txs += red[2][ww]; tys += red[3][ww];
      }
      total += Ss - (2.f / n) * ds + (txs * tys) / (n * n);
    }
    __syncthreads();
  }
  if (threadIdx.x == 0) out[0] = total / ((n - 1.f) * (n - 1.f));
}

// ---------------------------------------------------------------------------
extern "C" void kernel_launch(void* const* d_in, const int* in_sizes, int n_in,
                              void* d_out, int out_size, void* d_ws, size_t ws_size,
                              hipStream_t stream) {
  (void)in_sizes; (void)n_in; (void)out_size; (void)ws_size;
  const float* X = (const float*)d_in[0];   // (N, D1, L) fp32
  const float* Y = (const float*)d_in[1];   // (N, D2, L) fp32

  float* ws = (float*)d_ws;
  float* XT = ws;                                   // L*N*D
  float* YT = XT + (size_t)HL * HN * HD;            // L*N*D
  float* SQX = YT + (size_t)HL * HN * HD;           // L*N
  float* SQY = SQX + (size_t)HL * HN;               // L*N
  float* PART = SQY + (size_t)HL * HN;              // L*256*4

  hsic_prep<<<dim3((HN * HL) / 256), 256, 0, stream>>>(X, XT, SQX);
  hsic_prep<<<dim3((HN * HL) / 256), 256, 0, stream>>>(Y, YT, SQY);
  hsic_main<<<dim3(256, HL), 256, 0, stream>>>(XT, YT, SQX, SQY, PART);
  hsic_final<<<1, 256, 0, stream>>>(PART, (float*)d_out);
}